// ProtoSupConLoss_58901181497956
// MI455X (gfx1250) — compile-verified
//
#include <hip/hip_runtime.h>
#include <stdint.h>

typedef float v2f __attribute__((ext_vector_type(2)));
typedef float v8f __attribute__((ext_vector_type(8)));

#define B_  8
#define S_  1024
#define D_  768
#define C_  256
#define ASTRIDE 772   // 768 + 4 pad -> rows land on distinct LDS banks

// workspace layout (floats) -- the six stat arrays are contiguous on purpose
#define OFF_SQ      0                        // [B_*S_] feature row norms
#define OFF_CSQ     (OFF_SQ + B_*S_)         // [C_]    centroid norms
#define OFF_NEGSUM  (OFF_CSQ + C_)           // [B_*S_]  q=0
#define OFF_NEGCNT  (OFF_NEGSUM + B_*S_)     //          q=1
#define OFF_POSSUM  (OFF_NEGCNT + B_*S_)     //          q=2
#define OFF_POSCNT  (OFF_POSSUM + B_*S_)     //          q=3
#define OFF_CENTSUM (OFF_POSCNT + B_*S_)     //          q=4
#define OFF_POSCENT (OFF_CENTSUM + B_*S_)    //          q=5

// -------------------------------------------------------------------------
// Kernel 0: squared L2 norms of all feature rows (8192) + centroid rows (256).
// -------------------------------------------------------------------------
__global__ __launch_bounds__(256)
void row_norms_kernel(const float* __restrict__ F, const float* __restrict__ Cent,
                      float* __restrict__ ws) {
    int wave = threadIdx.x >> 5;
    int lane = threadIdx.x & 31;
    int row  = blockIdx.x * 8 + wave;                 // grid sized exactly: 8448 rows
    const float* p = (row < B_ * S_) ? (F + (size_t)row * D_)
                                     : (Cent + (size_t)(row - B_ * S_) * D_);
    float s = 0.f;
    for (int k = lane; k < D_; k += 32) { float x = p[k]; s += x * x; }
    for (int m = 16; m > 0; m >>= 1) s += __shfl_xor(s, m, 32);
    if (lane == 0) ws[row] = s;
}

// -------------------------------------------------------------------------
// Kernel 1 (fused): pairwise + centroid distance stats for one 16-row i-tile.
//  * A tile staged via global_load_async_to_lds_b128 (ASYNCcnt DMA path)
//  * Each wave processes TWO output tiles at a time sharing the A fragment
//  * Branchless masked accumulation (cndmask + fmac, no exec divergence)
// -------------------------------------------------------------------------
__global__ __launch_bounds__(128)
void stats_kernel(const float* __restrict__ F, const float* __restrict__ Cent,
                  const int* __restrict__ labels, const float* __restrict__ phi,
                  float* __restrict__ ws) {
    __shared__ float As[16 * ASTRIDE];
    __shared__ float red[4][6][16];

    const int b  = blockIdx.y;
    const int i0 = blockIdx.x * 16;
    const float* Fb = F + (size_t)b * S_ * D_;

    // ---- async DMA stage of the 16 x 768 A-tile into LDS (16B chunks) ----
    for (int idx = threadIdx.x; idx < 16 * (D_ / 4); idx += blockDim.x) {
        int r  = idx / (D_ / 4);
        int c4 = idx - r * (D_ / 4);
        const float* g = Fb + (size_t)(i0 + r) * D_ + c4 * 4;
        unsigned lds_off = (unsigned)(uintptr_t)&As[r * ASTRIDE + c4 * 4];
        asm volatile("global_load_async_to_lds_b128 %0, %1, off"
                     :: "v"(lds_off), "v"(g) : "memory");
    }
    asm volatile("s_wait_asynccnt 0x0" ::: "memory");
    __syncthreads();

    const int lane = threadIdx.x & 31;
    const int wave = threadIdx.x >> 5;
    const int half = lane >> 4;           // C rows 0..7 (half=0) / 8..15 (half=1)
    const int nsub = lane & 15;           // N index (and A/B fragment row)
    const int koff = half * 2;

    float sqi[8]; int labi[8];
    #pragma unroll
    for (int v = 0; v < 8; ++v) {
        int row = i0 + v + half * 8;
        sqi[v]  = ws[OFF_SQ + b * S_ + row];
        labi[v] = labels[b * S_ + row];
    }

    float negs[8] = {0,0,0,0,0,0,0,0}, negc[8] = {0,0,0,0,0,0,0,0};
    float poss[8] = {0,0,0,0,0,0,0,0}, posc[8] = {0,0,0,0,0,0,0,0};

    const float* Ap = &As[nsub * ASTRIDE + koff];

    // ---- pairwise tiles: each wave owns tile pairs {2w,2w+1}, stride 8 ----
    for (int jt = wave * 2; jt < S_ / 16; jt += 8) {
        const int j0 = jt * 16;
        const int j1 = j0 + 16;
        const float sqj0 = ws[OFF_SQ + b * S_ + j0 + nsub];
        const float sqj1 = ws[OFF_SQ + b * S_ + j1 + nsub];
        const int   lj0  = labels[b * S_ + j0 + nsub];
        const int   lj1  = labels[b * S_ + j1 + nsub];
        const float* Bp0 = Fb + (size_t)(j0 + nsub) * D_ + koff;
        const float* Bp1 = Fb + (size_t)(j1 + nsub) * D_ + koff;
        if (jt + 8 < S_ / 16)  // prefetch next pair's rows (global_prefetch_b8)
            __builtin_prefetch(Fb + (size_t)(j0 + 128 + nsub) * D_, 0, 0);

        v8f acc0 = {}, acc1 = {};
        #pragma unroll 4
        for (int k = 0; k < D_; k += 4) {
            v2f a, q0, q1;
            a.x  = Ap[k];   a.y  = Ap[k + 1];
            q0.x = Bp0[k];  q0.y = Bp0[k + 1];
            q1.x = Bp1[k];  q1.y = Bp1[k + 1];
            acc0 = __builtin_amdgcn_wmma_f32_16x16x4_f32(
                       false, a, false, q0, (short)0, acc0, false, false);
            acc1 = __builtin_amdgcn_wmma_f32_16x16x4_f32(
                       false, a, false, q1, (short)0, acc1, false, false);
        }

        #pragma unroll
        for (int v = 0; v < 8; ++v) {
            const int gi = i0 + v + half * 8;
            // tile 0
            float d2 = sqi[v] + sqj0 - 2.0f * acc0[v];
            float ds = (d2 > 0.f) ? sqrtf(d2) : 0.f;
            float nm = (lj0 != labi[v]) ? 1.f : 0.f;
            float pm = (lj0 == labi[v] && labi[v] >= 0 && gi != (j0 + nsub)) ? 1.f : 0.f;
            negs[v] = fmaf(ds, nm, negs[v]);  negc[v] += nm;
            poss[v] = fmaf(ds, pm, poss[v]);  posc[v] += pm;
            // tile 1
            d2 = sqi[v] + sqj1 - 2.0f * acc1[v];
            ds = (d2 > 0.f) ? sqrtf(d2) : 0.f;
            nm = (lj1 != labi[v]) ? 1.f : 0.f;
            pm = (lj1 == labi[v] && labi[v] >= 0 && gi != (j1 + nsub)) ? 1.f : 0.f;
            negs[v] = fmaf(ds, nm, negs[v]);  negc[v] += nm;
            poss[v] = fmaf(ds, pm, poss[v]);  posc[v] += pm;
        }
    }

    // ---- centroid tiles: wave owns pairs {2w,2w+1}, stride 8 (2 iters) ----
    float cents[8] = {0,0,0,0,0,0,0,0}, pcent[8] = {0,0,0,0,0,0,0,0};
    int kcl[8];
    #pragma unroll
    for (int v = 0; v < 8; ++v) kcl[v] = (labi[v] < 0) ? 0 : labi[v];

    for (int ct = wave * 2; ct < C_ / 16; ct += 8) {
        const int c0 = ct * 16;
        const int c1 = c0 + 16;
        const float csq0 = ws[OFF_CSQ + c0 + nsub];
        const float csq1 = ws[OFF_CSQ + c1 + nsub];
        const float ip0  = 1.0f / phi[c0 + nsub];
        const float ip1  = 1.0f / phi[c1 + nsub];
        const float* Bp0 = Cent + (size_t)(c0 + nsub) * D_ + koff;
        const float* Bp1 = Cent + (size_t)(c1 + nsub) * D_ + koff;

        v8f acc0 = {}, acc1 = {};
        #pragma unroll 4
        for (int k = 0; k < D_; k += 4) {
            v2f a, q0, q1;
            a.x  = Ap[k];   a.y  = Ap[k + 1];
            q0.x = Bp0[k];  q0.y = Bp0[k + 1];
            q1.x = Bp1[k];  q1.y = Bp1[k + 1];
            acc0 = __builtin_amdgcn_wmma_f32_16x16x4_f32(
                       false, a, false, q0, (short)0, acc0, false, false);
            acc1 = __builtin_amdgcn_wmma_f32_16x16x4_f32(
                       false, a, false, q1, (short)0, acc1, false, false);
        }

        #pragma unroll
        for (int v = 0; v < 8; ++v) {
            float d2 = sqi[v] + csq0 - 2.0f * acc0[v];
            float cd = ((d2 > 0.f) ? sqrtf(d2) : 0.f) * ip0;
            float sel = ((c0 + nsub) == kcl[v]) ? 1.f : 0.f;
            cents[v] += cd;
            pcent[v] = fmaf(cd, sel, pcent[v]);

            d2 = sqi[v] + csq1 - 2.0f * acc1[v];
            cd = ((d2 > 0.f) ? sqrtf(d2) : 0.f) * ip1;
            sel = ((c1 + nsub) == kcl[v]) ? 1.f : 0.f;
            cents[v] += cd;
            pcent[v] = fmaf(cd, sel, pcent[v]);
        }
    }

    // ---- reduce across the 16 lanes sharing the same M ----
    for (int m = 1; m < 16; m <<= 1) {
        #pragma unroll
        for (int v = 0; v < 8; ++v) {
            negs[v]  += __shfl_xor(negs[v],  m, 32);
            negc[v]  += __shfl_xor(negc[v],  m, 32);
            poss[v]  += __shfl_xor(poss[v],  m, 32);
            posc[v]  += __shfl_xor(posc[v],  m, 32);
            cents[v] += __shfl_xor(cents[v], m, 32);
            pcent[v] += __shfl_xor(pcent[v], m, 32);
        }
    }
    if (nsub == 0) {
        #pragma unroll
        for (int v = 0; v < 8; ++v) {
            int row = v + half * 8;
            red[wave][0][row] = negs[v];
            red[wave][1][row] = negc[v];
            red[wave][2][row] = poss[v];
            red[wave][3][row] = posc[v];
            red[wave][4][row] = cents[v];
            red[wave][5][row] = pcent[v];
        }
    }
    __syncthreads();
    if (threadIdx.x < 96) {
        int row = threadIdx.x & 15, q = threadIdx.x >> 4;   // q = 0..5
        float s = red[0][q][row] + red[1][q][row] + red[2][q][row] + red[3][q][row];
        ws[OFF_NEGSUM + q * (B_ * S_) + b * S_ + i0 + row] = s;
    }
}

// -------------------------------------------------------------------------
// Kernel 2: per-anchor loss terms + global mean (single block).
// -------------------------------------------------------------------------
__global__ __launch_bounds__(256)
void finalize_kernel(const float* __restrict__ ws, const int* __restrict__ labels,
                     const float* __restrict__ spk, float* __restrict__ out) {
    __shared__ float ssum[256], scnt[256];
    float lsum = 0.f, lcnt = 0.f;
    for (int idx = threadIdx.x; idx < B_ * S_; idx += 256) {
        int   lab   = labels[idx];
        float negcf = ws[OFF_NEGCNT + idx];
        float pos_x = ws[OFF_POSSUM + idx] / fmaxf(ws[OFF_POSCNT + idx], 1.f);
        float neg_s = ws[OFF_NEGSUM + idx] / fmaxf(negcf, 1.f);
        float pc    = ws[OFF_POSCENT + idx];
        float nc    = (ws[OFF_CENTSUM + idx] - pc) / (float)(C_ - 1);
        float ps    = 0.5f * pos_x + 0.5f * pc;       // SIG = 0.5
        float ns    = 0.5f * neg_s + 0.5f * nc;
        int   k     = (lab < 0) ? 0 : lab;
        float total = (ps - ns + 10.0f + expf(ps) - 1.0f) / spk[k];  // EPSILON = 10
        float anchor = (lab >= 0 && negcf > 0.f) ? 1.f : 0.f;
        lsum += total * anchor;
        lcnt += anchor;
    }
    ssum[threadIdx.x] = lsum;
    scnt[threadIdx.x] = lcnt;
    __syncthreads();
    for (int s = 128; s > 0; s >>= 1) {
        if (threadIdx.x < s) {
            ssum[threadIdx.x] += ssum[threadIdx.x + s];
            scnt[threadIdx.x] += scnt[threadIdx.x + s];
        }
        __syncthreads();
    }
    if (threadIdx.x == 0) out[0] = ssum[0] / fmaxf(scnt[0], 1.f);
}

extern "C" void kernel_launch(void* const* d_in, const int* in_sizes, int n_in,
                              void* d_out, int out_size, void* d_ws, size_t ws_size,
                              hipStream_t stream) {
    const float* features  = (const float*)d_in[0];   // [8,1024,768] f32
    const int*   labels    = (const int*)  d_in[1];   // [8,1024] i32
    const float* centroids = (const float*)d_in[2];   // [256,768] f32
    const float* phi       = (const float*)d_in[3];   // [256] f32
    const float* spk       = (const float*)d_in[4];   // [256] f32
    float* ws  = (float*)d_ws;
    float* out = (float*)d_out;

    row_norms_kernel<<<dim3((B_ * S_ + C_) / 8), 256, 0, stream>>>(features, centroids, ws);
    stats_kernel    <<<dim3(S_ / 16, B_),        128, 0, stream>>>(features, centroids, labels, phi, ws);
    finalize_kernel <<<1,                         256, 0, stream>>>(ws, labels, spk, out);
}